// CAViaR_43808666419435
// MI455X (gfx1250) — compile-verified
//
#include <hip/hip_runtime.h>

typedef __attribute__((ext_vector_type(2))) float v2f;
typedef __attribute__((ext_vector_type(8))) float v8f;

#define T_STEPS 2048
#define BATCH   128
#define HDIM    100
#define G4      400      // 4*H gate rows
#define NW      25       // waves = row tiles of 16
#define NK      25       // K chunks of 4 (K = 100)
#define NTHREADS (NW * 32)

__device__ __forceinline__ float fsig(float x)  { return 1.0f / (1.0f + __expf(-x)); }
__device__ __forceinline__ float ftanh(float x) { return 1.0f - 2.0f / (1.0f + __expf(2.0f * x)); }

__global__ __launch_bounds__(NTHREADS, 1)
void lstm_b0_wmma_kernel(const float* __restrict__ x,     // [T,B,1]
                         const float* __restrict__ W_ih,  // [4H,1]
                         const float* __restrict__ W_hh,  // [4H,H]
                         const float* __restrict__ b_ih,  // [4H]
                         const float* __restrict__ b_hh,  // [4H]
                         const float* __restrict__ W1,    // [64,H]
                         const float* __restrict__ b1,    // [64]
                         const float* __restrict__ W2,    // [1,64]
                         const float* __restrict__ b2,    // [1]
                         float* __restrict__ out)         // [1]
{
    // LDS: h state (+ zero slot for inactive WMMA-B lanes), gates, staged inputs, head reduce
    __shared__ float s_h[104];        // [0..99]=h, [100..101]=zeros (B-matrix filler), 8B aligned
    __shared__ float s_gates[G4];
    __shared__ float s_x[T_STEPS];    // batch-0 inputs
    __shared__ float s_red[64];

    const int tid  = threadIdx.x;
    const int lane = tid & 31;
    const int wave = tid >> 5;        // 0..24 -> gate rows [16w, 16w+16)

    // ---- one-time staging -------------------------------------------------
    for (int t = tid; t < T_STEPS; t += NTHREADS) s_x[t] = x[t * BATCH]; // x[t,0,0]
    if (tid < 104) s_h[tid] = 0.0f;                                      // h0 = 0, zero slot
    if (tid < G4)  s_gates[tid] = 0.0f;

    // A-matrix (W_hh) resident in VGPRs, WMMA 16x16x4 f32 A layout:
    //   lanes 0-15 : M = lane,     K = {4c+0, 4c+1}
    //   lanes16-31 : M = lane-16,  K = {4c+2, 4c+3}
    v2f a[NK];
    {
        const int row  = 16 * wave + (lane & 15);
        const int koff = (lane >> 4) * 2;
        const float* wrow = W_hh + row * HDIM + koff;
        #pragma unroll
        for (int c = 0; c < NK; ++c) {
            a[c].x = wrow[4 * c + 0];
            a[c].y = wrow[4 * c + 1];
        }
    }

    // Per-thread recurrent-cell constants for the elementwise phase (threads 0..99)
    float wi0 = 0.f, wi1 = 0.f, wi2 = 0.f, wi3 = 0.f;
    float bb0 = 0.f, bb1 = 0.f, bb2 = 0.f, bb3 = 0.f;
    float cst = 0.f;                                   // cell state c, register-resident
    if (tid < HDIM) {
        wi0 = W_ih[tid];            bb0 = b_ih[tid]          + b_hh[tid];
        wi1 = W_ih[HDIM + tid];     bb1 = b_ih[HDIM + tid]   + b_hh[HDIM + tid];
        wi2 = W_ih[2*HDIM + tid];   bb2 = b_ih[2*HDIM + tid] + b_hh[2*HDIM + tid];
        wi3 = W_ih[3*HDIM + tid];   bb3 = b_ih[3*HDIM + tid] + b_hh[3*HDIM + tid];
    }

    // B-matrix gather plan: column 0 of B carries h, all other columns zero.
    //   lane 0  -> {h[4c],   h[4c+1]}   (B rows K=0,1 of chunk c)
    //   lane 16 -> {h[4c+2], h[4c+3]}   (B rows K=2,3 of chunk c)
    //   others  -> zero slot
    const v2f* hb = (const v2f*)s_h;
    int bidx, bstep;
    if      (lane == 0)  { bidx = 0;  bstep = 2; }
    else if (lane == 16) { bidx = 1;  bstep = 2; }
    else                 { bidx = 50; bstep = 0; }     // floats 100..101 == zeros

    __syncthreads();

    // ---- 2048-step recurrence --------------------------------------------
    for (int t = 0; t < T_STEPS; ++t) {
        // gates(:,0) = W_hh @ h  via 25 chained f32 WMMAs (exact f32 accumulate)
        v8f acc = {};
        int bi = bidx;
        #pragma unroll
        for (int c = 0; c < NK; ++c) {
            v2f b = hb[bi];
            bi += bstep;
            acc = __builtin_amdgcn_wmma_f32_16x16x4_f32(
                false, a[c], false, b, (short)0, acc, false, false);
        }
        // D column 0: lane 0 holds rows 16w..16w+7 (acc[0..7]), lane 16 rows 16w+8..16w+15
        if (lane == 0) {
            #pragma unroll
            for (int i = 0; i < 8; ++i) s_gates[16 * wave + i] = acc[i];
        } else if (lane == 16) {
            #pragma unroll
            for (int i = 0; i < 8; ++i) s_gates[16 * wave + 8 + i] = acc[i];
        }
        __syncthreads();

        if (tid < HDIM) {
            const float xt = s_x[t];
            const float gi = s_gates[tid]            + xt * wi0 + bb0;
            const float gf = s_gates[HDIM   + tid]   + xt * wi1 + bb1;
            const float gg = s_gates[2*HDIM + tid]   + xt * wi2 + bb2;
            const float go = s_gates[3*HDIM + tid]   + xt * wi3 + bb3;
            const float ig = fsig(gi);
            const float fg = fsig(gf);
            const float gv = ftanh(gg);
            const float og = fsig(go);
            cst = fg * cst + ig * gv;
            s_h[tid] = og * ftanh(cst);
        }
        __syncthreads();
    }

    // ---- head: out = W2 @ (W1 @ h_{T-1} + b1) + b2  (scalar) --------------
    if (tid < 64) {
        float l = b1[tid];
        #pragma unroll 4
        for (int j = 0; j < HDIM; ++j) l += W1[tid * HDIM + j] * s_h[j];
        s_red[tid] = l * W2[tid];
    }
    __syncthreads();
    if (tid == 0) {
        float o = b2[0];
        #pragma unroll
        for (int m = 0; m < 64; ++m) o += s_red[m];
        out[0] = o;
    }
}

extern "C" void kernel_launch(void* const* d_in, const int* in_sizes, int n_in,
                              void* d_out, int out_size, void* d_ws, size_t ws_size,
                              hipStream_t stream) {
    (void)in_sizes; (void)n_in; (void)out_size; (void)d_ws; (void)ws_size;
    const float* x    = (const float*)d_in[0];
    const float* W_ih = (const float*)d_in[1];
    const float* W_hh = (const float*)d_in[2];
    const float* b_ih = (const float*)d_in[3];
    const float* b_hh = (const float*)d_in[4];
    const float* W1   = (const float*)d_in[5];
    const float* b1   = (const float*)d_in[6];
    const float* W2   = (const float*)d_in[7];
    const float* b2   = (const float*)d_in[8];
    float* out        = (float*)d_out;

    lstm_b0_wmma_kernel<<<dim3(1), dim3(NTHREADS), 0, stream>>>(
        x, W_ih, W_hh, b_ih, b_hh, W1, b1, W2, b2, out);
}